// GaussianRasterizer_72816875536912
// MI455X (gfx1250) — compile-verified
//
#include <hip/hip_runtime.h>
#include <math.h>

#define NG  1024
#define IMH 128
#define IMW 128

typedef float v2f __attribute__((ext_vector_type(2)));
typedef float v8f __attribute__((ext_vector_type(8)));

// ---------------------------------------------------------------------------
// Kernel 1: per-gaussian preprocessing.
// Emits per gaussian: 8 quadratic-form coefficients (K-padded to 8 for WMMA),
// color+masked-opacity (float4), and depth key.
// power(px,py) = c0*px^2 + c1*py^2 + c2*px*py + c3*px + c4*py + c5
// ---------------------------------------------------------------------------
__global__ void __launch_bounds__(256)
preprocess_kernel(const float* __restrict__ means3D,
                  const float* __restrict__ sh,
                  const float* __restrict__ opacities,
                  const float* __restrict__ scales,
                  const float* __restrict__ rotations,
                  const float* __restrict__ cam_pos,
                  const float* __restrict__ cam_rot,
                  const float* __restrict__ thf_p,
                  const float* __restrict__ near_far,
                  float*  __restrict__ coeff,     // [NG][8]
                  float4* __restrict__ colop,     // [NG] rgb + masked opacity
                  float*  __restrict__ keys)      // [NG] depth
{
    int n = blockIdx.x * blockDim.x + threadIdx.x;
    if (n >= NG) return;

    const float thf = thf_p[0];
    const float fx  = (float)IMW / (2.0f * thf);
    const float fy  = (float)IMH / (2.0f * thf);

    float R[9];
#pragma unroll
    for (int i = 0; i < 9; ++i) R[i] = cam_rot[i];

    float m0 = means3D[n*3+0] - cam_pos[0];
    float m1 = means3D[n*3+1] - cam_pos[1];
    float m2 = means3D[n*3+2] - cam_pos[2];

    float t0 = R[0]*m0 + R[1]*m1 + R[2]*m2;
    float t1 = R[3]*m0 + R[4]*m1 + R[5]*m2;
    float t2 = R[6]*m0 + R[7]*m1 + R[8]*m2;

    float lim = 1.3f * thf;
    float txc = fminf(fmaxf(t0 / t2, -lim), lim) * t2;
    float tyc = fminf(fmaxf(t1 / t2, -lim), lim) * t2;

    // quaternion -> rotation
    float qr = rotations[n*4+0], qx = rotations[n*4+1];
    float qy = rotations[n*4+2], qz = rotations[n*4+3];
    float qn = rsqrtf(qr*qr + qx*qx + qy*qy + qz*qz);
    qr *= qn; qx *= qn; qy *= qn; qz *= qn;
    float Rq[9] = {
        1.0f-2.0f*(qy*qy+qz*qz), 2.0f*(qx*qy-qr*qz),      2.0f*(qx*qz+qr*qy),
        2.0f*(qx*qy+qr*qz),      1.0f-2.0f*(qx*qx+qz*qz), 2.0f*(qy*qz-qr*qx),
        2.0f*(qx*qz-qr*qy),      2.0f*(qy*qz+qr*qx),      1.0f-2.0f*(qx*qx+qy*qy)};

    float s[3] = { scales[n*3+0], scales[n*3+1], scales[n*3+2] };
    float M[9];
#pragma unroll
    for (int i = 0; i < 3; ++i)
#pragma unroll
        for (int j = 0; j < 3; ++j) M[i*3+j] = Rq[i*3+j] * s[j];

    float cov3[9];
#pragma unroll
    for (int i = 0; i < 3; ++i)
#pragma unroll
        for (int k = 0; k < 3; ++k)
            cov3[i*3+k] = M[i*3+0]*M[k*3+0] + M[i*3+1]*M[k*3+1] + M[i*3+2]*M[k*3+2];

    float z2  = t2 * t2;
    float J00 = fx / t2, J02 = -fx * txc / z2;
    float J11 = fy / t2, J12 = -fy * tyc / z2;

    float T0[3], T1[3];
#pragma unroll
    for (int k = 0; k < 3; ++k) {
        T0[k] = J00 * R[0*3+k] + J02 * R[2*3+k];
        T1[k] = J11 * R[1*3+k] + J12 * R[2*3+k];
    }
    float u0[3], u1[3];
#pragma unroll
    for (int i = 0; i < 3; ++i) {
        u0[i] = cov3[i*3+0]*T0[0] + cov3[i*3+1]*T0[1] + cov3[i*3+2]*T0[2];
        u1[i] = cov3[i*3+0]*T1[0] + cov3[i*3+1]*T1[1] + cov3[i*3+2]*T1[2];
    }
    float a = T0[0]*u0[0] + T0[1]*u0[1] + T0[2]*u0[2] + 0.3f;
    float b = T0[0]*u1[0] + T0[1]*u1[1] + T0[2]*u1[2];
    float c = T1[0]*u1[0] + T1[1]*u1[1] + T1[2]*u1[2] + 0.3f;

    float det = a*c - b*b;
    float inv_det = 1.0f / det;
    float conA =  c * inv_det;
    float conB = -b * inv_det;
    float conC =  a * inv_det;

    float mx = fx * t0 / t2 + ((float)IMW - 1.0f) * 0.5f;
    float my = fy * t1 / t2 + ((float)IMH - 1.0f) * 0.5f;

    // SH -> color (degree 3)
    float dn = rsqrtf(m0*m0 + m1*m1 + m2*m2);
    float x = m0*dn, y = m1*dn, z = m2*dn;
    float xx = x*x, yy = y*y, zz = z*z;
    float xy = x*y, yz = y*z, xz = x*z;
    float col[3];
#pragma unroll
    for (int ch = 0; ch < 3; ++ch) {
        const float* S = sh + n*48 + ch;   // stride 3 over k
        float r = 0.28209479177387814f * S[0*3];
        r += -0.4886025119029199f * y * S[1*3]
           +  0.4886025119029199f * z * S[2*3]
           -  0.4886025119029199f * x * S[3*3];
        r +=  1.0925484305920792f  * xy * S[4*3]
           + -1.0925484305920792f  * yz * S[5*3]
           +  0.31539156525252005f * (2.0f*zz - xx - yy) * S[6*3]
           + -1.0925484305920792f  * xz * S[7*3]
           +  0.5462742152960396f  * (xx - yy) * S[8*3];
        r += -0.5900435899266435f * y * (3.0f*xx - yy) * S[9*3]
           +  2.890611442640554f  * xy * z * S[10*3]
           + -0.4570457994644658f * y * (4.0f*zz - xx - yy) * S[11*3]
           +  0.3731763325901154f * z * (2.0f*zz - 3.0f*xx - 3.0f*yy) * S[12*3]
           + -0.4570457994644658f * x * (4.0f*zz - xx - yy) * S[13*3]
           +  1.445305721320277f  * z * (xx - yy) * S[14*3]
           + -0.5900435899266435f * x * (xx - 3.0f*yy) * S[15*3];
        col[ch] = fmaxf(r + 0.5f, 0.0f);
    }

    bool valid = (t2 > near_far[0]) && (t2 < near_far[1]) && (det > 0.0f);

    coeff[n*8+0] = -0.5f * conA;
    coeff[n*8+1] = -0.5f * conC;
    coeff[n*8+2] = -conB;
    coeff[n*8+3] = conA * mx + conB * my;
    coeff[n*8+4] = conC * my + conB * mx;
    coeff[n*8+5] = -0.5f * (conA*mx*mx + conC*my*my) - conB*mx*my;
    coeff[n*8+6] = 0.0f;
    coeff[n*8+7] = 0.0f;

    colop[n] = make_float4(col[0], col[1], col[2], valid ? opacities[n] : 0.0f);
    keys[n]  = t2;
}

// ---------------------------------------------------------------------------
// Kernel 2: one-block bitonic depth sort of 1024 gaussians + gather into
// depth-ordered coefficient / color arrays.
// ---------------------------------------------------------------------------
__global__ void __launch_bounds__(1024)
sort_gather_kernel(const float* __restrict__ keys,
                   const float* __restrict__ coeff,
                   const float4* __restrict__ colop,
                   float*  __restrict__ sortedCoeff,
                   float4* __restrict__ sortedColOp)
{
    __shared__ float sk[NG];
    __shared__ int   si[NG];
    int tid = threadIdx.x;
    sk[tid] = keys[tid];
    si[tid] = tid;
    __syncthreads();

    for (int k = 2; k <= NG; k <<= 1) {
        for (int j = k >> 1; j > 0; j >>= 1) {
            int ixj = tid ^ j;
            if (ixj > tid) {
                bool asc = ((tid & k) == 0);
                float ka = sk[tid], kb = sk[ixj];
                if ((ka > kb) == asc) {
                    sk[tid] = kb; sk[ixj] = ka;
                    int t = si[tid]; si[tid] = si[ixj]; si[ixj] = t;
                }
            }
            __syncthreads();
        }
    }

    int src = si[tid];
#pragma unroll
    for (int q = 0; q < 8; ++q)
        sortedCoeff[tid*8+q] = coeff[src*8+q];
    sortedColOp[tid] = colop[src];
}

// ---------------------------------------------------------------------------
// Kernel 3: tiled render. Block = 256 threads (8 wave32) = 16x8 pixel tile.
// Each wave owns one 16-pixel row. Per 16-gaussian chunk, the quadratic-form
// powers are computed as a K=8 f32 WMMA (two 16x16x4 accumulations):
//   D[g, p] = sum_k coeff_k[g] * feat_k[p]   (M=gaussian, N=pixel)
// D layout puts lane p <-> pixel p; lane p blends gaussians 0-7 of the chunk,
// lane p+16 blends 8-15; a shfl_xor(16) merge keeps strict depth order.
// ---------------------------------------------------------------------------
__global__ void __launch_bounds__(256)
render_kernel(const float*  __restrict__ gCoeff,
              const float4* __restrict__ gColOp,
              const float*  __restrict__ bg_color,
              float* __restrict__ out)
{
    __shared__ float  sCoeff[NG * 8];   // 32 KB
    __shared__ float4 sColOp[NG];       // 16 KB

    int tid = threadIdx.x;

    // Cooperative LDS staging (uniform trip counts -> no divergence).
    const float4* src4 = (const float4*)gCoeff;
    float4* dst4 = (float4*)sCoeff;
    for (int i = tid; i < NG * 2; i += 256) dst4[i] = src4[i];
    for (int i = tid; i < NG; i += 256)     sColOp[i] = gColOp[i];
    __syncthreads();

    int lane = tid & 31;
    int wave = tid >> 5;
    int p    = lane & 15;
    bool hi  = lane >= 16;

    int x = blockIdx.x * 16 + p;
    int y = blockIdx.y * 8 + wave;
    float px = (float)x, py = (float)y;

    // B operand (pixel features), constant over the chunk loop.
    // K layout: lane<16 holds K={0,1} in {b.x,b.y}; lane>=16 holds K={2,3}.
    v2f b1 = hi ? (v2f){px * py, px}      : (v2f){px * px, py * py};
    v2f b2 = hi ? (v2f){0.0f, 0.0f}       : (v2f){py, 1.0f};

    float accT = 1.0f;
    float accR = 0.0f, accG = 0.0f, accB = 0.0f;

    for (int chunk = 0; chunk < NG / 16; ++chunk) {
        // A operand: 16 gaussian coefficient rows (M = lane&15).
        const float* cp = &sCoeff[(chunk * 16 + p) * 8];
        int off = hi ? 2 : 0;
        v2f a1 = { cp[off + 0], cp[off + 1] };
        v2f a2 = { cp[off + 4], cp[off + 5] };

        v8f d = {};
        d = __builtin_amdgcn_wmma_f32_16x16x4_f32(false, a1, false, b1,
                                                  (short)0, d, false, false);
        d = __builtin_amdgcn_wmma_f32_16x16x4_f32(false, a2, false, b2,
                                                  (short)0, d, false, false);

        // Sequential blend of this lane's 8 gaussians (strict depth order).
        float segT = 1.0f;
        float segR = 0.0f, segG = 0.0f, segB = 0.0f;
        int gofs = chunk * 16 + (hi ? 8 : 0);
#pragma unroll
        for (int r = 0; r < 8; ++r) {
            float  pw = d[r];
            float4 co = sColOp[gofs + r];
            float  al = fminf(co.w * __expf(pw), 0.99f);
            al = (pw <= 0.0f && al >= (1.0f / 255.0f)) ? al : 0.0f;
            float w = al * segT;
            segR += w * co.x; segG += w * co.y; segB += w * co.z;
            segT *= (1.0f - al);
        }

        // Merge low half (gaussians 0-7) with high half (8-15).
        float oT = __shfl_xor(segT, 16);
        float oR = __shfl_xor(segR, 16);
        float oG = __shfl_xor(segG, 16);
        float oB = __shfl_xor(segB, 16);
        float cT = segT * oT;
        float cR, cG, cB;
        if (!hi) { cR = segR + segT * oR; cG = segG + segT * oG; cB = segB + segT * oB; }
        else     { cR = oR + oT * segR;   cG = oG + oT * segG;   cB = oB + oT * segB;   }

        accR += accT * cR;
        accG += accT * cG;
        accB += accT * cB;
        accT *= cT;
    }

    if (!hi) {
        int pix = y * IMW + x;
        out[0 * IMH * IMW + pix] = accR + accT * bg_color[0];
        out[1 * IMH * IMW + pix] = accG + accT * bg_color[1];
        out[2 * IMH * IMW + pix] = accB + accT * bg_color[2];
    }
}

// ---------------------------------------------------------------------------
extern "C" void kernel_launch(void* const* d_in, const int* in_sizes, int n_in,
                              void* d_out, int out_size, void* d_ws, size_t ws_size,
                              hipStream_t stream)
{
    const float* means3D   = (const float*)d_in[0];
    const float* sh        = (const float*)d_in[1];
    const float* opacities = (const float*)d_in[2];
    const float* scales    = (const float*)d_in[3];
    const float* rotations = (const float*)d_in[4];
    const float* cam_pos   = (const float*)d_in[5];
    const float* cam_rot   = (const float*)d_in[6];
    const float* thf       = (const float*)d_in[7];
    const float* bg_color  = (const float*)d_in[8];
    const float* near_far  = (const float*)d_in[9];
    float* out = (float*)d_out;

    char* ws = (char*)d_ws;
    float*  coeff       = (float*)(ws + 0);        // 32 KB
    float4* colop       = (float4*)(ws + 32768);   // 16 KB
    float*  keys        = (float*)(ws + 49152);    //  4 KB
    float*  sortedCoeff = (float*)(ws + 53248);    // 32 KB
    float4* sortedColOp = (float4*)(ws + 86016);   // 16 KB  (total 100 KB)

    preprocess_kernel<<<NG / 256, 256, 0, stream>>>(
        means3D, sh, opacities, scales, rotations, cam_pos, cam_rot, thf,
        near_far, coeff, colop, keys);

    sort_gather_kernel<<<1, NG, 0, stream>>>(
        keys, coeff, colop, sortedCoeff, sortedColOp);

    render_kernel<<<dim3(IMW / 16, IMH / 8), 256, 0, stream>>>(
        sortedCoeff, sortedColOp, bg_color, out);
}